// GAT_64845416235490
// MI455X (gfx1250) — compile-verified
//
#include <hip/hip_runtime.h>
#include <hip/hip_bf16.h>
#include <limits.h>

// ---------------------------------------------------------------------------
// GAT (2-layer) for MI455X / gfx1250.  FP32 WMMA (v_wmma_f32_16x16x4_f32) for
// both GEMMs (fully branch-free inner loop, compile-time strides, 2 N-tiles
// per wave for A reuse); edge softmax via monotone-int atomicMax + f32
// atomicAdd; message scatter one wave32 per edge.
// ---------------------------------------------------------------------------

#define N_NODES   50000
#define N_EDGES   800000
#define IN_DIM    256
#define HIDDEN    256
#define HEADS     8
#define FPH       32
#define NCLS      40
#define NCLS_PAD  48
#define NEG_SLOPE 0.2f

typedef float v2f __attribute__((ext_vector_type(2)));
typedef float v8f __attribute__((ext_vector_type(8)));

// ---------------- monotone float<->int encoding for atomicMax ---------------
__device__ __forceinline__ int enc_f32(float f) {
    int i = __float_as_int(f);
    return (i < 0) ? (i ^ 0x7fffffff) : i;
}
__device__ __forceinline__ float dec_f32(int i) {
    return __int_as_float((i < 0) ? (i ^ 0x7fffffff) : i);
}

// ---------------------------------------------------------------------------
// FP32 WMMA GEMM:  C[M,NOUT] = A[M,K] * B[K,NB]  (row-major, NB >= NOUT tiles)
// One wave computes NTILES 16x16 C tiles (cols n0 and n0+TSTRIDE) sharing one
// A fragment.  blockIdx.x = M tile (M % 16 == 0).  All strides compile-time so
// every load is an immediate-offset global_load; no branches in the K loop.
// ---------------------------------------------------------------------------
template<int K, int NB, int NOUT, int NTILES, int TSTRIDE>
__global__ void gat_wmma_gemm(const float* __restrict__ A,
                              const float* __restrict__ B,
                              float* __restrict__ C) {
    const int lane  = threadIdx.x & 31;
    const int wave  = threadIdx.x >> 5;
    const int m0    = blockIdx.x << 4;
    const int row   = lane & 15;          // A row within tile
    const int col   = lane & 15;          // B/C column within tile
    const int khalf = (lane >> 4) << 1;   // 0 (lanes 0-15) or 2 (lanes 16-31)

    const float* __restrict__ Arow = A + (size_t)(m0 + row) * K + khalf;

    const v8f vzero = {};
    v8f acc[NTILES];
    const float* __restrict__ Bp[NTILES];
#pragma unroll
    for (int j = 0; j < NTILES; ++j) {
        acc[j] = vzero;
        Bp[j]  = B + (wave << 4) + j * TSTRIDE + col + (size_t)khalf * NB;
    }

#pragma unroll 8
    for (int k = 0; k < K; k += 4) {
        // A fragment 16x4: lane holds A[m][khalf], A[m][khalf+1] (8B aligned)
        v2f a = *(const v2f*)(Arow + k);
#pragma unroll
        for (int j = 0; j < NTILES; ++j) {
            // B fragment 4x16: VGPR0 = row (k+khalf), VGPR1 = row (k+khalf+1)
            v2f b;
            b.x = Bp[j][(size_t)k * NB];
            b.y = Bp[j][(size_t)k * NB + NB];
            acc[j] = __builtin_amdgcn_wmma_f32_16x16x4_f32(
                         false, a, false, b, (short)0, acc[j], false, false);
        }
    }

    // C layout: VGPR v -> row (v + 8*(lane>=16)), col = lane&15
    const int mbase = m0 + ((lane >> 4) << 3);
#pragma unroll
    for (int j = 0; j < NTILES; ++j) {
        const int n = (wave << 4) + j * TSTRIDE + col;
        if ((NOUT & 15) == 0 || n < NOUT) {
            float* __restrict__ Crow = C + n;
#pragma unroll
            for (int v = 0; v < 8; ++v)
                Crow[(size_t)(mbase + v) * NOUT] = acc[j][v];
        }
    }
}

// ---------------- fills / init ---------------------------------------------
__global__ void gat_fill_f32(float* __restrict__ p, float v, int n) {
    int t = blockIdx.x * blockDim.x + threadIdx.x;
    if (t < n) p[t] = v;
}
__global__ void gat_fill_i32(int* __restrict__ p, int v, int n) {
    int t = blockIdx.x * blockDim.x + threadIdx.x;
    if (t < n) p[t] = v;
}
__global__ void gat_init_out_bias(float* __restrict__ out,
                                  const float* __restrict__ b2, int n) {
    int t = blockIdx.x * blockDim.x + threadIdx.x;
    if (t < n) out[t] = b2[t % NCLS];
}
// zero-pad W2 [256,40] -> W2p [256,48] so GEMM2's K loop is branch-free
__global__ void gat_pad_w2(const float* __restrict__ W2, float* __restrict__ W2p) {
    int t = blockIdx.x * blockDim.x + threadIdx.x;
    if (t >= HIDDEN * NCLS_PAD) return;
    int k = t / NCLS_PAD, n = t - k * NCLS_PAD;
    W2p[t] = (n < NCLS) ? W2[k * NCLS + n] : 0.f;
}

// ---------------- per-node attention dot products --------------------------
__global__ void gat_alpha_l1(const float* __restrict__ h,
                             const float* __restrict__ a_src,
                             const float* __restrict__ a_dst,
                             float* __restrict__ as, float* __restrict__ ad) {
    int t = blockIdx.x * blockDim.x + threadIdx.x;
    if (t >= N_NODES * HEADS) return;
    const int hd = t & (HEADS - 1);
    const float* __restrict__ hp = h + (size_t)t * FPH;
    const float* __restrict__ sp = a_src + hd * FPH;
    const float* __restrict__ dp = a_dst + hd * FPH;
    float accs = 0.f, accd = 0.f;
#pragma unroll
    for (int f = 0; f < FPH; ++f) {
        float v = hp[f];
        accs += v * sp[f];
        accd += v * dp[f];
    }
    as[t] = accs;
    ad[t] = accd;
}
__global__ void gat_alpha_l2(const float* __restrict__ h,
                             const float* __restrict__ a_src,
                             const float* __restrict__ a_dst,
                             float* __restrict__ as, float* __restrict__ ad) {
    int t = blockIdx.x * blockDim.x + threadIdx.x;
    if (t >= N_NODES) return;
    const float* __restrict__ hp = h + (size_t)t * NCLS;
    float accs = 0.f, accd = 0.f;
#pragma unroll
    for (int f = 0; f < NCLS; ++f) {
        float v = hp[f];
        accs += v * a_src[f];
        accd += v * a_dst[f];
    }
    as[t] = accs;
    ad[t] = accd;
}

// ---------------- edge pass A: logits + segment max ------------------------
template<int H>
__global__ void gat_edge_logit_max(const int* __restrict__ src,
                                   const int* __restrict__ dst,
                                   const float* __restrict__ as,
                                   const float* __restrict__ ad,
                                   float* __restrict__ elog,
                                   int* __restrict__ mx) {
    int t = blockIdx.x * blockDim.x + threadIdx.x;
    if (t >= N_EDGES * H) return;
    const int e  = t / H;
    const int hd = t - e * H;
    const float v0 = as[src[e] * H + hd] + ad[dst[e] * H + hd];
    const float v  = (v0 > 0.f) ? v0 : NEG_SLOPE * v0;
    elog[t] = v;
    atomicMax(&mx[dst[e] * H + hd], enc_f32(v));
}

// ---------------- edge pass B: exp + segment sum ---------------------------
template<int H>
__global__ void gat_edge_exp_sum(const int* __restrict__ dst,
                                 float* __restrict__ elog,   // in: logit, out: exp
                                 const int* __restrict__ mx,
                                 float* __restrict__ sum) {
    int t = blockIdx.x * blockDim.x + threadIdx.x;
    if (t >= N_EDGES * H) return;
    const int e  = t / H;
    const int hd = t - e * H;
    const int si = dst[e] * H + hd;
    const float ex = __expf(elog[t] - dec_f32(mx[si]));
    elog[t] = ex;
    atomicAdd(&sum[si], ex);
}

// ---------------- edge pass C: message scatter (layer 1) -------------------
// One wave32 per edge; lane handles 8 contiguous features (head = lane/4).
__global__ void gat_scatter_l1(const int* __restrict__ src,
                               const int* __restrict__ dst,
                               const float* __restrict__ ex,
                               const float* __restrict__ sum,
                               const float* __restrict__ h,
                               float* __restrict__ out) {
    const int wid  = (blockIdx.x * blockDim.x + threadIdx.x) >> 5;
    const int lane = threadIdx.x & 31;
    if (wid >= N_EDGES) return;
    const int sN = src[wid], dN = dst[wid];
    const int hd = lane >> 2;                      // 4 lanes per head
    const float a = ex[(size_t)wid * HEADS + hd] /
                    (sum[(size_t)dN * HEADS + hd] + 1e-16f);
    const float4* __restrict__ hp =
        (const float4*)(h + (size_t)sN * HIDDEN + lane * 8);
    float* __restrict__ op = out + (size_t)dN * HIDDEN + lane * 8;
    float4 v0 = hp[0], v1 = hp[1];
    atomicAdd(op + 0, v0.x * a); atomicAdd(op + 1, v0.y * a);
    atomicAdd(op + 2, v0.z * a); atomicAdd(op + 3, v0.w * a);
    atomicAdd(op + 4, v1.x * a); atomicAdd(op + 5, v1.y * a);
    atomicAdd(op + 6, v1.z * a); atomicAdd(op + 7, v1.w * a);
}

// ---------------- edge pass C: message scatter (layer 2, H=1, F=40) --------
__global__ void gat_scatter_l2(const int* __restrict__ src,
                               const int* __restrict__ dst,
                               const float* __restrict__ ex,
                               const float* __restrict__ sum,
                               const float* __restrict__ h,
                               float* __restrict__ out) {
    const int wid  = (blockIdx.x * blockDim.x + threadIdx.x) >> 5;
    const int lane = threadIdx.x & 31;
    if (wid >= N_EDGES) return;
    const int sN = src[wid], dN = dst[wid];
    const float a = ex[wid] / (sum[dN] + 1e-16f);
    const float* __restrict__ hp = h + (size_t)sN * NCLS;
    float* __restrict__ op = out + (size_t)dN * NCLS;
    atomicAdd(op + lane, hp[lane] * a);                       // f = 0..31
    if (lane < NCLS - 32)
        atomicAdd(op + 32 + lane, hp[32 + lane] * a);         // f = 32..39
}

// ---------------- bias + ELU (layer-1 epilogue) ----------------------------
__global__ void gat_bias_elu(const float* __restrict__ acc,
                             const float* __restrict__ b,
                             float* __restrict__ out, int n) {
    int t = blockIdx.x * blockDim.x + threadIdx.x;
    if (t >= n) return;
    float v = acc[t] + b[t & (HIDDEN - 1)];
    out[t] = (v > 0.f) ? v : (__expf(v) - 1.f);
}

// ---------------------------------------------------------------------------
extern "C" void kernel_launch(void* const* d_in, const int* in_sizes, int n_in,
                              void* d_out, int out_size, void* d_ws, size_t ws_size,
                              hipStream_t stream) {
    const float* x      = (const float*)d_in[0];
    const int*   ei     = (const int*)  d_in[1];
    const float* W1     = (const float*)d_in[2];
    const float* a_src1 = (const float*)d_in[3];
    const float* a_dst1 = (const float*)d_in[4];
    const float* b1     = (const float*)d_in[5];
    const float* W2     = (const float*)d_in[6];
    const float* a_src2 = (const float*)d_in[7];
    const float* a_dst2 = (const float*)d_in[8];
    const float* b2     = (const float*)d_in[9];
    const int* src = ei;
    const int* dst = ei + N_EDGES;
    float* out = (float*)d_out;

    // -------- workspace layout (floats); heavy buffers are reused ----------
    float* base = (float*)d_ws;
    float* h1    = base;                       // [N, 256]; reused as h_elu
    float* out1  = h1   + (size_t)N_NODES * HIDDEN;   // accum; reused as h2
    float* elog  = out1 + (size_t)N_NODES * HIDDEN;   // [E, 8]; l2 reuses [E]
    float* as1   = elog + (size_t)N_EDGES * HEADS;    // [N, 8]
    float* ad1   = as1  + (size_t)N_NODES * HEADS;
    int*   m1    = (int*)(ad1 + (size_t)N_NODES * HEADS);
    float* s1    = (float*)m1 + (size_t)N_NODES * HEADS;
    float* as2   = s1   + (size_t)N_NODES * HEADS;    // [N]
    float* ad2   = as2  + N_NODES;
    int*   m2    = (int*)(ad2 + N_NODES);
    float* s2    = (float*)m2 + N_NODES;
    float* W2p   = s2 + N_NODES;               // [256, 48] zero-padded W2

    const int TB = 256;
    const int mtiles = N_NODES / 16;           // 3125 (exact)
    const int EH1 = N_EDGES * HEADS;
    const int NH1 = N_NODES * HEADS;

    // ---------------- layer 1 ---------------------------------------------
    // h1 = x @ W1 : 8 waves/block, 2 tiles/wave (n0 and n0+128) -> 256 cols
    gat_wmma_gemm<IN_DIM, HIDDEN, HIDDEN, 2, 128>
        <<<mtiles, 256, 0, stream>>>(x, W1, h1);
    gat_alpha_l1<<<(NH1 + TB - 1) / TB, TB, 0, stream>>>(h1, a_src1, a_dst1, as1, ad1);
    gat_fill_i32<<<(NH1 + TB - 1) / TB, TB, 0, stream>>>(m1, INT_MIN, NH1);
    gat_fill_f32<<<(NH1 + TB - 1) / TB, TB, 0, stream>>>(s1, 0.f, NH1);
    gat_fill_f32<<<(N_NODES * HIDDEN + TB - 1) / TB, TB, 0, stream>>>(out1, 0.f, N_NODES * HIDDEN);
    gat_edge_logit_max<HEADS><<<(EH1 + TB - 1) / TB, TB, 0, stream>>>(src, dst, as1, ad1, elog, m1);
    gat_edge_exp_sum<HEADS><<<(EH1 + TB - 1) / TB, TB, 0, stream>>>(dst, elog, m1, s1);
    gat_scatter_l1<<<N_EDGES / 8, 256, 0, stream>>>(src, dst, elog, s1, h1, out1);
    // h_elu = elu(out1 + b1)  (overwrites h1; h1 dead after scatter)
    gat_bias_elu<<<(N_NODES * HIDDEN + TB - 1) / TB, TB, 0, stream>>>(out1, b1, h1, N_NODES * HIDDEN);

    // ---------------- layer 2 ---------------------------------------------
    float* h2 = out1;                          // reuse accumulator buffer
    gat_pad_w2<<<(HIDDEN * NCLS_PAD + TB - 1) / TB, TB, 0, stream>>>(W2, W2p);
    // h2 = h_elu @ W2p : 3 waves/block cover 48 padded cols, store 40
    gat_wmma_gemm<HIDDEN, NCLS_PAD, NCLS, 1, 0>
        <<<mtiles, 96, 0, stream>>>(h1, W2p, h2);
    gat_alpha_l2<<<(N_NODES + TB - 1) / TB, TB, 0, stream>>>(h2, a_src2, a_dst2, as2, ad2);
    gat_fill_i32<<<(N_NODES + TB - 1) / TB, TB, 0, stream>>>(m2, INT_MIN, N_NODES);
    gat_fill_f32<<<(N_NODES + TB - 1) / TB, TB, 0, stream>>>(s2, 0.f, N_NODES);
    gat_init_out_bias<<<(N_NODES * NCLS + TB - 1) / TB, TB, 0, stream>>>(out, b2, N_NODES * NCLS);
    gat_edge_logit_max<1><<<(N_EDGES + TB - 1) / TB, TB, 0, stream>>>(src, dst, as2, ad2, elog, m2);
    gat_edge_exp_sum<1><<<(N_EDGES + TB - 1) / TB, TB, 0, stream>>>(dst, elog, m2, s2);
    gat_scatter_l2<<<N_EDGES / 8, 256, 0, stream>>>(src, dst, elog, s2, h2, out);
}